// GCNReg_print_29703993819342
// MI455X (gfx1250) — compile-verified
//
#include <hip/hip_runtime.h>

typedef __attribute__((ext_vector_type(2))) float v2f;
typedef __attribute__((ext_vector_type(8))) float v8f;

#define NN   100000     // nodes
#define NE   1600000    // edges
#define IND  64
#define HID  128
#define NG   512        // graphs

// ---------------------------------------------------------------------------
// Degree histogram: deg_out[src]++, deg_in[dst]++
// ---------------------------------------------------------------------------
__global__ void gcn_deg_kernel(const int* __restrict__ src, const int* __restrict__ dst,
                               float* __restrict__ deg_out, float* __restrict__ deg_in) {
    int e = blockIdx.x * blockDim.x + threadIdx.x;
    if (e < NE) {
        atomicAdd(&deg_out[src[e]], 1.0f);
        atomicAdd(&deg_in[dst[e]], 1.0f);
    }
}

// In-place d[i] = rsqrt(max(d[i], 1))
__global__ void gcn_rsqrt_kernel(float* __restrict__ d, int n) {
    int i = blockIdx.x * blockDim.x + threadIdx.x;
    if (i < n) d[i] = rsqrtf(fmaxf(d[i], 1.0f));
}

// ---------------------------------------------------------------------------
// Fused src-norm + scatter-add:  agg[dst][d] += h[src][d] * rs_out[src]
// Each thread handles 4 consecutive dims of one edge (float4 gather, 4 atomics).
// dim_shift = log2(dim); dim in {64,128}. All index math fits in int32.
// ---------------------------------------------------------------------------
__global__ void gcn_edge_agg_kernel(const float* __restrict__ h, const float* __restrict__ rs,
                                    const int* __restrict__ src, const int* __restrict__ dst,
                                    float* __restrict__ agg, int dim_shift) {
    int idx = blockIdx.x * blockDim.x + threadIdx.x;
    int qshift = dim_shift - 2;                 // groups of 4 floats per edge
    int e = idx >> qshift;
    if (e >= NE) return;
    int q = idx & ((1 << qshift) - 1);
    int s = src[e];
    int d = dst[e];
    float scale = rs[s];
    const float4 hv = *(const float4*)(h + (s << dim_shift) + (q << 2));
    float* ap = agg + (d << dim_shift) + (q << 2);
    atomicAdd(ap + 0, hv.x * scale);
    atomicAdd(ap + 1, hv.y * scale);
    atomicAdd(ap + 2, hv.z * scale);
    atomicAdd(ap + 3, hv.w * scale);
}

// ---------------------------------------------------------------------------
// WMMA fp32 GEMM: out[M,N] = act( (row_scale .* A[M,K]) @ W[K,N] + bias )
// One wave (32 lanes) computes a 16x16 tile with V_WMMA_F32_16X16X4_F32.
// A fragment: lane l<16 -> A[row][4k+0..1], lane l>=16 -> A[row][4k+2..3]
// B fragment: lane l<16 -> W[4k+0..1][col], lane l>=16 -> W[4k+2..3][col]
// D: VGPR v -> row base+v+8*half, col base+(lane&15)
// M, K, N assumed multiples of 16 / 4 / 16 (true for this problem).
// ---------------------------------------------------------------------------
__device__ __forceinline__ v8f gcn_wmma_tile(const float* __restrict__ A,
                                             const float* __restrict__ W,
                                             const float* __restrict__ row_scale,
                                             int row, int col, int K, int N, int half) {
    float rs = row_scale ? row_scale[row] : 1.0f;
    v8f acc = {};
    const float* arow = A + (long long)row * K;
    for (int k0 = 2 * half; k0 < K; k0 += 4) {
        v2f a = *(const v2f*)(arow + k0);       // 8B-aligned (k0 even)
        a.x *= rs;
        a.y *= rs;
        v2f b;
        b.x = W[(long long)k0 * N + col];
        b.y = W[(long long)(k0 + 1) * N + col];
        acc = __builtin_amdgcn_wmma_f32_16x16x4_f32(
            /*neg_a=*/false, a, /*neg_b=*/false, b,
            /*c_mod=*/(short)0, acc, /*reuse_a=*/false, /*reuse_b=*/false);
    }
    return acc;
}

__global__ void gcn_wmma_gemm_kernel(const float* __restrict__ A, const float* __restrict__ W,
                                     const float* __restrict__ bias,
                                     const float* __restrict__ row_scale,
                                     float* __restrict__ out,
                                     int M, int K, int N, int apply_relu) {
    int lane = threadIdx.x;                     // 0..31 (wave32)
    int l = lane & 15;
    int half = lane >> 4;
    int rowbase = blockIdx.x * 16;
    int colbase = blockIdx.y * 16;
    int row = rowbase + l;                      // A-fragment row for this lane
    int col = colbase + l;                      // B/D column for this lane
    (void)M;
    v8f acc = gcn_wmma_tile(A, W, row_scale, row, col, K, N, half);
    float bv = bias[col];
#pragma unroll
    for (int v = 0; v < 8; ++v) {
        int r = rowbase + v + 8 * half;
        float val = acc[v] + bv;
        if (apply_relu) val = fmaxf(val, 0.0f);
        out[(long long)r * N + col] = val;
    }
}

// Layer-2 variant: ReLU output is pooled directly:
//   sums[graph_ids[r]][col] += relu(tile + bias)
__global__ void gcn_wmma_gemm_pool_kernel(const float* __restrict__ A, const float* __restrict__ W,
                                          const float* __restrict__ bias,
                                          const float* __restrict__ row_scale,
                                          const int* __restrict__ graph_ids,
                                          float* __restrict__ sums,
                                          int K, int N) {
    int lane = threadIdx.x;
    int l = lane & 15;
    int half = lane >> 4;
    int rowbase = blockIdx.x * 16;
    int colbase = blockIdx.y * 16;
    int row = rowbase + l;
    int col = colbase + l;
    v8f acc = gcn_wmma_tile(A, W, row_scale, row, col, K, N, half);
    float bv = bias[col];
#pragma unroll
    for (int v = 0; v < 8; ++v) {
        int r = rowbase + v + 8 * half;
        float val = fmaxf(acc[v] + bv, 0.0f);
        atomicAdd(&sums[graph_ids[r] * N + col], val);
    }
}

// ---------------------------------------------------------------------------
// Per-graph node counts, then hg = sums / max(cnt,1)
// ---------------------------------------------------------------------------
__global__ void gcn_graph_count_kernel(const int* __restrict__ gid, float* __restrict__ cnts) {
    int i = blockIdx.x * blockDim.x + threadIdx.x;
    if (i < NN) atomicAdd(&cnts[gid[i]], 1.0f);
}

__global__ void gcn_mean_div_kernel(const float* __restrict__ sums, const float* __restrict__ cnts,
                                    float* __restrict__ hg) {
    int i = blockIdx.x * blockDim.x + threadIdx.x;
    if (i < NG * HID) hg[i] = sums[i] / fmaxf(cnts[i >> 7], 1.0f);
}

// Final 128 -> 1 projection: out[g] = dot(h[g,:], Wc3[:,0]) + bc3[0]
__global__ void gcn_final_dot_kernel(const float* __restrict__ h, const float* __restrict__ Wc3,
                                     const float* __restrict__ bc3, float* __restrict__ out) {
    __shared__ float red[HID];
    int g = blockIdx.x;
    int t = threadIdx.x;
    red[t] = h[g * HID + t] * Wc3[t];
    __syncthreads();
#pragma unroll
    for (int s = HID / 2; s > 0; s >>= 1) {
        if (t < s) red[t] += red[t + s];
        __syncthreads();
    }
    if (t == 0) out[g] = red[0] + bc3[0];
}

// ---------------------------------------------------------------------------
// Host launcher
// ---------------------------------------------------------------------------
extern "C" void kernel_launch(void* const* d_in, const int* in_sizes, int n_in,
                              void* d_out, int out_size, void* d_ws, size_t ws_size,
                              hipStream_t stream) {
    (void)in_sizes; (void)n_in; (void)out_size; (void)ws_size;

    const float* x    = (const float*)d_in[0];
    const int*   esrc = (const int*)  d_in[1];
    const int*   edst = (const int*)  d_in[2];
    const int*   gid  = (const int*)  d_in[3];
    const float* W1   = (const float*)d_in[4];
    const float* b1   = (const float*)d_in[5];
    const float* W2   = (const float*)d_in[6];
    const float* b2   = (const float*)d_in[7];
    const float* Wc1  = (const float*)d_in[8];
    const float* bc1  = (const float*)d_in[9];
    const float* Wc2  = (const float*)d_in[10];
    const float* bc2  = (const float*)d_in[11];
    const float* Wc3  = (const float*)d_in[12];
    const float* bc3  = (const float*)d_in[13];

    // Workspace layout (floats): ~130 MB total
    float* ws      = (float*)d_ws;
    float* rs_out  = ws;                      // NN  (deg_out -> rsqrt, in place)
    float* rs_in   = rs_out + NN;             // NN
    float* agg1    = rs_in  + NN;             // NN*64
    float* h1      = agg1   + (size_t)NN * IND;   // NN*128
    float* agg2    = h1     + (size_t)NN * HID;   // NN*128
    float* sums    = agg2   + (size_t)NN * HID;   // NG*HID
    float* cnts    = sums   + NG * HID;           // NG
    float* hg      = cnts   + NG;                 // NG*HID
    float* m1      = hg     + NG * HID;           // NG*HID
    float* m2      = m1     + NG * HID;           // NG*HID

    // Zero the accumulation buffers every call (deterministic).
    hipMemsetAsync(rs_out, 0, (size_t)2 * NN * sizeof(float), stream);
    hipMemsetAsync(agg1,   0, (size_t)NN * IND * sizeof(float), stream);
    hipMemsetAsync(agg2,   0, (size_t)NN * HID * sizeof(float), stream);
    hipMemsetAsync(sums,   0, (size_t)(NG * HID + NG) * sizeof(float), stream);

    const int TB = 256;

    // Degrees -> rsqrt norms
    gcn_deg_kernel<<<(NE + TB - 1) / TB, TB, 0, stream>>>(esrc, edst, rs_out, rs_in);
    gcn_rsqrt_kernel<<<(2 * NN + TB - 1) / TB, TB, 0, stream>>>(rs_out, 2 * NN);

    // Layer 1: agg1[dst] += x[src]*rs_out[src];  h1 = relu(rs_in.*agg1 @ W1 + b1)
    {
        long long thr = (long long)NE * (IND / 4);
        gcn_edge_agg_kernel<<<(unsigned)((thr + TB - 1) / TB), TB, 0, stream>>>(
            x, rs_out, esrc, edst, agg1, 6);
        dim3 grid(NN / 16, HID / 16);   // 6250 x 8
        gcn_wmma_gemm_kernel<<<grid, 32, 0, stream>>>(agg1, W1, b1, rs_in, h1,
                                                      NN, IND, HID, 1);
    }

    // Layer 2: agg2[dst] += h1[src]*rs_out[src];  pooled relu GEMM -> sums
    {
        long long thr = (long long)NE * (HID / 4);
        gcn_edge_agg_kernel<<<(unsigned)((thr + TB - 1) / TB), TB, 0, stream>>>(
            h1, rs_out, esrc, edst, agg2, 7);
        dim3 grid(NN / 16, HID / 16);
        gcn_wmma_gemm_pool_kernel<<<grid, 32, 0, stream>>>(agg2, W2, b2, rs_in,
                                                           gid, sums, HID, HID);
    }

    // Mean pooling
    gcn_graph_count_kernel<<<(NN + TB - 1) / TB, TB, 0, stream>>>(gid, cnts);
    gcn_mean_div_kernel<<<(NG * HID + TB - 1) / TB, TB, 0, stream>>>(sums, cnts, hg);

    // MLP head
    {
        dim3 grid(NG / 16, HID / 16);   // 32 x 8
        gcn_wmma_gemm_kernel<<<grid, 32, 0, stream>>>(hg, Wc1, bc1, nullptr, m1,
                                                      NG, HID, HID, 1);
        gcn_wmma_gemm_kernel<<<grid, 32, 0, stream>>>(m1, Wc2, bc2, nullptr, m2,
                                                      NG, HID, HID, 1);
    }
    gcn_final_dot_kernel<<<NG, HID, 0, stream>>>(m2, Wc3, bc3, (float*)d_out);
}